// GCN_75359496175833
// MI455X (gfx1250) — compile-verified
//
#include <hip/hip_runtime.h>

typedef float v2f __attribute__((ext_vector_type(2)));
typedef float v8f __attribute__((ext_vector_type(8)));

#define NNODES 100000
#define INDIM  128
#define HIDDIM 128
#define OUTDIM 70
#define OUTPAD 80

__device__ __forceinline__ void atomAddF32(float* p, float v) {
    unsafeAtomicAdd(p, v);   // -> global_atomic_add_f32 on gfx1250
}

// ---------------- degree / normalization ----------------
__global__ void k_init_deg(float* deg) {
    int i = blockIdx.x * blockDim.x + threadIdx.x;
    if (i < NNODES) deg[i] = 1.0f;            // self-loop contributes 1
}

__global__ void k_count_deg(const long long* __restrict__ dst, float* deg, int E) {
    int e = blockIdx.x * blockDim.x + threadIdx.x;
    if (e < E) atomAddF32(&deg[(int)dst[e]], 1.0f);
}

__global__ void k_dinv(float* deg) {
    int i = blockIdx.x * blockDim.x + threadIdx.x;
    if (i < NNODES) deg[i] = rsqrtf(deg[i]);  // deg >= 1 always
}

// ---------------- GEMM1: xw = x @ W1 (100000x128 @ 128x128), f32 WMMA ----------------
__global__ void __launch_bounds__(256) k_gemm1(const float* __restrict__ X,
                                               const float* __restrict__ W,
                                               float* __restrict__ XW) {
    __shared__ float sW[64 * 128];   // K-chunk of W1 (32 KB)
    __shared__ float sX[16 * 128];   // 16-row x tile (8 KB)
    const int tid  = threadIdx.x;
    const int row0 = blockIdx.x * 16;          // 6250 * 16 == 100000 exactly

    { // stage x tile: 2048 floats = 512 float4
        const float4* src = (const float4*)(X + (long long)row0 * INDIM);
        float4* dst = (float4*)sX;
        dst[tid]       = src[tid];
        dst[tid + 256] = src[tid + 256];
    }

    const int wave = tid >> 5;
    const int lane = tid & 31;
    const int rl   = lane & 15;                // M (for A) / N (for B) index
    const int kh   = (lane < 16) ? 0 : 2;      // half-wave K offset
    const int n0   = wave * 16;                // 8 waves cover N = 0..127
    v8f acc = {};

    for (int kc = 0; kc < 128; kc += 64) {
        __syncthreads();
        { // stage W1 rows kc..kc+63 : 8192 floats = 2048 float4
            const float4* wsrc = (const float4*)(W + kc * HIDDIM);
            float4* wdst = (float4*)sW;
#pragma unroll
            for (int i = 0; i < 8; ++i) wdst[tid + i * 256] = wsrc[tid + i * 256];
        }
        __syncthreads();
#pragma unroll
        for (int k = 0; k < 64; k += 4) {
            v2f a, b;
            a.x = sX[rl * 128 + kc + k + kh];
            a.y = sX[rl * 128 + kc + k + kh + 1];
            b.x = sW[(k + kh) * 128 + n0 + rl];
            b.y = sW[(k + kh + 1) * 128 + n0 + rl];
            acc = __builtin_amdgcn_wmma_f32_16x16x4_f32(false, a, false, b,
                                                        (short)0, acc, false, false);
        }
    }

    const int mo = (lane < 16) ? 0 : 8;
#pragma unroll
    for (int r = 0; r < 8; ++r)
        XW[(long long)(row0 + r + mo) * HIDDIM + n0 + rl] = acc[r];
}

// ---------------- GEMM2: hw = relu(h) @ W2 (128x70, N padded to 80) ----------------
__global__ void __launch_bounds__(256) k_gemm2(const float* __restrict__ H,
                                               const float* __restrict__ W2,
                                               float* __restrict__ HW) {
    __shared__ float sW[128 * OUTPAD];  // 40 KB, zero-padded cols 70..79
    __shared__ float sX[16 * 128];      // 8 KB
    const int tid  = threadIdx.x;
    const int row0 = blockIdx.x * 16;

    for (int i = tid; i < 16 * 128; i += 256)          // fused ReLU
        sX[i] = fmaxf(H[(long long)row0 * 128 + i], 0.0f);
    for (int i = tid; i < 128 * OUTPAD; i += 256) {
        int k = i / OUTPAD, n = i % OUTPAD;
        sW[i] = (n < OUTDIM) ? W2[k * OUTDIM + n] : 0.0f;
    }
    __syncthreads();

    const int wave = tid >> 5;
    const int lane = tid & 31;
    if (wave < 5) {                                    // wave-uniform: EXEC all-1 inside
        const int rl = lane & 15;
        const int kh = (lane < 16) ? 0 : 2;
        const int n0 = wave * 16;
        v8f acc = {};
#pragma unroll
        for (int k = 0; k < 128; k += 4) {
            v2f a, b;
            a.x = sX[rl * 128 + k + kh];
            a.y = sX[rl * 128 + k + kh + 1];
            b.x = sW[(k + kh) * OUTPAD + n0 + rl];
            b.y = sW[(k + kh + 1) * OUTPAD + n0 + rl];
            acc = __builtin_amdgcn_wmma_f32_16x16x4_f32(false, a, false, b,
                                                        (short)0, acc, false, false);
        }
        const int mo  = (lane < 16) ? 0 : 8;
        const int col = n0 + rl;
        if (col < OUTDIM) {
#pragma unroll
            for (int r = 0; r < 8; ++r)
                HW[(long long)(row0 + r + mo) * OUTDIM + col] = acc[r];
        }
    }
}

// ---------------- self-loop + bias init: out = xw * dinv^2 + b ----------------
template <int D>
__global__ void k_self(const float* __restrict__ XW, const float* __restrict__ dinv,
                       const float* __restrict__ bias, float* __restrict__ out, int total) {
    int idx = blockIdx.x * blockDim.x + threadIdx.x;
    if (idx >= total) return;
    int i = idx / D, d = idx % D;
    float di = dinv[i];
    out[idx] = XW[idx] * di * di + bias[d];
}

// ---------------- edge scatter, D=128: one wave per edge, float4 lanes ----------------
__global__ void __launch_bounds__(256) k_edge128(const long long* __restrict__ src,
                                                 const long long* __restrict__ dst,
                                                 const float* __restrict__ dinv,
                                                 const float* __restrict__ XW,
                                                 float* __restrict__ out, int E) {
    int w = (blockIdx.x * 256 + threadIdx.x) >> 5;
    if (w >= E) return;
    int lane = threadIdx.x & 31;
    int s = (int)src[w], t = (int)dst[w];
    float nrm = dinv[s] * dinv[t];
    float4 v = ((const float4*)(XW + (long long)s * 128))[lane];
    float* o = out + (long long)t * 128 + lane * 4;
    atomAddF32(o + 0, v.x * nrm);
    atomAddF32(o + 1, v.y * nrm);
    atomAddF32(o + 2, v.z * nrm);
    atomAddF32(o + 3, v.w * nrm);
}

// ---------------- edge scatter, D=70 ----------------
__global__ void __launch_bounds__(256) k_edge70(const long long* __restrict__ src,
                                                const long long* __restrict__ dst,
                                                const float* __restrict__ dinv,
                                                const float* __restrict__ XW,
                                                float* __restrict__ out, int E) {
    int w = (blockIdx.x * 256 + threadIdx.x) >> 5;
    if (w >= E) return;
    int lane = threadIdx.x & 31;
    int s = (int)src[w], t = (int)dst[w];
    float nrm = dinv[s] * dinv[t];
    const float* xs = XW + (long long)s * OUTDIM;
    float* o = out + (long long)t * OUTDIM;
    for (int d = lane; d < OUTDIM; d += 32)
        atomAddF32(o + d, xs[d] * nrm);
}

extern "C" void kernel_launch(void* const* d_in, const int* in_sizes, int n_in,
                              void* d_out, int out_size, void* d_ws, size_t ws_size,
                              hipStream_t stream) {
    const float*     x   = (const float*)d_in[0];
    const long long* ei  = (const long long*)d_in[1];   // int64 [2, E]
    const float*     W1  = (const float*)d_in[2];
    const float*     b1  = (const float*)d_in[3];
    const float*     W2  = (const float*)d_in[4];
    const float*     b2  = (const float*)d_in[5];
    float*           out = (float*)d_out;

    const int E = in_sizes[1] / 2;
    const long long* srcp = ei;
    const long long* dstp = ei + E;

    // workspace layout (all 16B aligned): deg/dinv | xw (reused as hw) | h
    char*  ws   = (char*)d_ws;
    float* deg  = (float*)ws;                                   // 400 KB
    float* xw   = (float*)(ws + (1 << 20));                     // 51.2 MB
    float* h    = (float*)(ws + (1 << 20) + 51200000);          // 51.2 MB
    float* hw   = xw;                                           // xw dead by GEMM2

    const int T = 256;
    const int nodeBlk  = (NNODES + T - 1) / T;
    const int edgeBlk  = (E + T - 1) / T;
    const int waveBlk  = (E * 32 + T - 1) / T;                  // one wave per edge

    // degrees + rsqrt normalization (in place)
    k_init_deg<<<nodeBlk, T, 0, stream>>>(deg);
    k_count_deg<<<edgeBlk, T, 0, stream>>>(dstp, deg, E);
    k_dinv<<<nodeBlk, T, 0, stream>>>(deg);

    // layer 1
    k_gemm1<<<NNODES / 16, T, 0, stream>>>(x, W1, xw);
    k_self<HIDDIM><<<(NNODES * HIDDIM + T - 1) / T, T, 0, stream>>>(xw, deg, b1, h,
                                                                    NNODES * HIDDIM);
    k_edge128<<<waveBlk, T, 0, stream>>>(srcp, dstp, deg, xw, h, E);

    // layer 2 (ReLU fused into GEMM2 tile load)
    k_gemm2<<<NNODES / 16, T, 0, stream>>>(h, W2, hw);
    k_self<OUTDIM><<<(NNODES * OUTDIM + T - 1) / T, T, 0, stream>>>(hw, deg, b2, out,
                                                                    NNODES * OUTDIM);
    k_edge70<<<waveBlk, T, 0, stream>>>(srcp, dstp, deg, hw, out, E);
}